// IDM_Layer_10806137716829
// MI455X (gfx1250) — compile-verified
//
#include <hip/hip_runtime.h>
#include <math.h>

#define STEPS 50
#define DT_C 0.1f
#define EPS_C 1e-5f
#define BLOCK 256
#define WAVE_VEH 64                       // vehicles per wave (2 per lane)
#define BLOCK_VEH ((BLOCK / 32) * WAVE_VEH)  // 512 vehicles per block

// Two vehicles per thread (independent chains -> VOPD/v_pk dual-issue).
// Each wave stages a 64x50 fp32 output tile in LDS (conflict-free banking),
// then drains it with 25 exactly-uniform CDNA5 async store-from-lds b128 ops
// using immediate offsets and non-temporal hints (streaming 200MB output).
__global__ __launch_bounds__(BLOCK) void idm_forward_kernel(
    const float* __restrict__ vi_g,
    const float* __restrict__ dv_g,
    const float* __restrict__ dd_g,
    const float* __restrict__ vf_p,
    const float* __restrict__ A_p,
    const float* __restrict__ b_p,
    const float* __restrict__ s0_p,
    const float* __restrict__ T_p,
    float* __restrict__ out,
    int n)
{
    __shared__ __align__(16) float tile[BLOCK_VEH * STEPS];  // 100 KB

    const int lane  = threadIdx.x & 31;
    const int wave  = threadIdx.x >> 5;
    const int wbase = blockIdx.x * BLOCK_VEH + wave * WAVE_VEH;
    const int i0    = wbase + lane;        // vehicle A
    const int i1    = i0 + 32;             // vehicle B

    // Learned scalar IDM parameters (single-element device arrays; uniform ->
    // compiler lowers to s_load + scalar-float ops, once per wave).
    const float vf = *vf_p;
    const float A  = *A_p;
    const float b  = *b_p;
    const float s0 = *s0_p;
    const float T  = *T_p;

    const float inv_2sqrtAb = 1.0f / (2.0f * sqrtf(A * b));
    const float inv_vf      = 1.0f / vf;

    float* wtile = tile + wave * (WAVE_VEH * STEPS);  // this wave's 12800 B tile

    if (i1 < n) {
        // Fast path: both vehicles valid. Two independent recurrences per lane.
        float va = vi_g[i0], vb = vi_g[i1];
        float da = dv_g[i0], db = dv_g[i1];
        const float ia = 1.0f / (dd_g[i0] + EPS_C);
        const float ib = 1.0f / (dd_g[i1] + EPS_C);
        float* ta = wtile + lane * STEPS;
        float* tb = wtile + (lane + 32) * STEPS;
#pragma unroll
        for (int s = 0; s < STEPS; ++s) {
            float sa = s0 + fmaxf(0.0f, fmaf(va * da, inv_2sqrtAb, va * T));
            float sb = s0 + fmaxf(0.0f, fmaf(vb * db, inv_2sqrtAb, vb * T));
            float xa = va * inv_vf,  xb = vb * inv_vf;
            float xa2 = xa * xa,     xb2 = xb * xb;
            float xa4 = xa2 * xa2,   xb4 = xb2 * xb2;
            float qa = sa * ia,      qb = sb * ib;
            float aha = A * (1.0f - xa4 - qa * qa);
            float ahb = A * (1.0f - xb4 - qb * qb);
            va = fmaf(aha,  DT_C, va);   vb = fmaf(ahb,  DT_C, vb);
            da = fmaf(aha, -DT_C, da);   db = fmaf(ahb, -DT_C, db);
            ta[s] = aha;
            tb[s] = ahb;
        }
    } else if (i0 < n) {
        // Tail lane: only vehicle A valid.
        float v  = vi_g[i0];
        float dv = dv_g[i0];
        const float inv_d = 1.0f / (dd_g[i0] + EPS_C);
        float* t = wtile + lane * STEPS;
#pragma unroll
        for (int s = 0; s < STEPS; ++s) {
            float sstar = s0 + fmaxf(0.0f, fmaf(v * dv, inv_2sqrtAb, v * T));
            float x  = v * inv_vf;
            float x2 = x * x;
            float x4 = x2 * x2;
            float q  = sstar * inv_d;
            float ah = A * (1.0f - x4 - q * q);
            v  = fmaf(ah,  DT_C, v);
            dv = fmaf(ah, -DT_C, dv);
            t[s] = ah;
        }
    }

    if (wbase >= n) return;  // fully idle wave

    if (wbase + WAVE_VEH <= n) {
        // Full wave: tile == out[wbase*50 .. wbase*50+3200) contiguously.
        // Wait for this wave's LDS writes before the async engine reads them.
        asm volatile("s_wait_dscnt 0x0" ::: "memory");

        // Low 32 bits of a generic shared pointer == LDS byte offset.
        // 64*50*4 B = 800 b128 chunks = exactly 25 per lane.
        // IOFFSET is applied to BOTH the global and LDS addresses, so one base
        // address pair per lane + compile-time offsets covers the whole drain.
        const unsigned la = (unsigned)(size_t)wtile + (unsigned)lane * 16u;
        const unsigned long long ga =
            (unsigned long long)(size_t)(out + (size_t)wbase * STEPS) +
            (unsigned long long)lane * 16ull;
#pragma unroll
        for (int k = 0; k < 25; ++k) {
            asm volatile(
                "global_store_async_from_lds_b128 %0, %1, off offset:%2 th:TH_STORE_NT"
                :
                : "v"(ga), "v"(la), "i"(k * 512)
                : "memory");
        }
        asm volatile("s_wait_asynccnt 0x0" ::: "memory");
    } else {
        // Partial tail wave: plain per-vehicle drain of the staged rows.
        if (i0 < n) {
            const float* t = wtile + lane * STEPS;
            for (int s = 0; s < STEPS; ++s)
                out[(size_t)i0 * STEPS + s] = t[s];
        }
        if (i1 < n) {
            const float* t = wtile + (lane + 32) * STEPS;
            for (int s = 0; s < STEPS; ++s)
                out[(size_t)i1 * STEPS + s] = t[s];
        }
    }
}

extern "C" void kernel_launch(void* const* d_in, const int* in_sizes, int n_in,
                              void* d_out, int out_size, void* d_ws, size_t ws_size,
                              hipStream_t stream) {
    (void)n_in; (void)out_size; (void)d_ws; (void)ws_size;
    const float* vi = (const float*)d_in[0];
    const float* dv = (const float*)d_in[1];
    const float* dd = (const float*)d_in[2];
    const float* vf = (const float*)d_in[3];
    const float* A  = (const float*)d_in[4];
    const float* b  = (const float*)d_in[5];
    const float* s0 = (const float*)d_in[6];
    const float* T  = (const float*)d_in[7];
    const int n     = in_sizes[0];
    const int grid  = (n + BLOCK_VEH - 1) / BLOCK_VEH;
    idm_forward_kernel<<<grid, BLOCK, 0, stream>>>(vi, dv, dd, vf, A, b, s0, T,
                                                   (float*)d_out, n);
}